// TinySAGE_38869454028882
// MI455X (gfx1250) — compile-verified
//
#include <hip/hip_runtime.h>
#include <stdint.h>

// CDNA5 fp32 WMMA fragments (wave32):
//  A 16x4 f32  -> 2 VGPRs/lane : lanes 0-15 hold rows M=0..15 {K=k0,k0+1}; lanes 16-31 {K=k0+2,k0+3}
//  B 4x16 f32  -> 2 VGPRs/lane : lanes 0-15 hold cols N=0..15 {K=k0,k0+1}; lanes 16-31 {K=k0+2,k0+3}
//  C/D 16x16   -> 8 VGPRs/lane : VGPR j holds M=j (lanes 0-15) / M=8+j (lanes 16-31), N = lane&15
typedef float v2f __attribute__((ext_vector_type(2)));
typedef float v8f __attribute__((ext_vector_type(8)));

// ---------------------------------------------------------------------------
// Scatter-add of feat[src] into agg[dst], one thread per (edge, channel).
// C = 1<<Cshift channels. Optionally accumulates float degree (c==0 lanes).
// Feature reads for one edge are 128B/256B-coalesced; atomic targets for one
// edge are contiguous dwords -> resolved as L2-resident f32 atomics (agg
// buffers are 12.8/25.6 MB, well inside the 192 MB L2).
// ---------------------------------------------------------------------------
__global__ __launch_bounds__(256)
void sage_scatter(const long long* __restrict__ ei,   // [2,E] int64 (src row, dst row)
                  const float*     __restrict__ feat, // [N, C]
                  float*           __restrict__ agg,  // [N, C] (pre-zeroed)
                  float*           __restrict__ deg,  // [N] or nullptr (pre-zeroed)
                  long long E, int Cshift)
{
    long long t = (long long)blockIdx.x * 256 + threadIdx.x;
    long long total = E << Cshift;
    if (t >= total) return;
    long long e = t >> Cshift;
    int       c = (int)(t & ((1 << Cshift) - 1));
    long long s = ei[e];
    long long d = ei[E + e];
    atomicAdd(&agg[((size_t)d << Cshift) + c], feat[((size_t)s << Cshift) + c]);
    if (deg && c == 0) atomicAdd(&deg[d], 1.0f);
}

// ---------------------------------------------------------------------------
// Layer 1 node update: h = relu( (agg1/deg) @ W_l + b + x @ W_r ), K=32 -> 64
// One wave per 16-node tile; 4 column tiles of 16; 8 K-steps of 2 fp32 WMMAs.
// All exits/guards around the WMMA region are wave-uniform (EXEC all-1s).
// ---------------------------------------------------------------------------
__global__ __launch_bounds__(256)
void sage_node_l1(const float* __restrict__ x,    // [N,32]
                  const float* __restrict__ agg,  // [N,32] (sums)
                  const float* __restrict__ deg,  // [N]
                  const float* __restrict__ Wl,   // [32,64] row-major (K,N)
                  const float* __restrict__ Wr,   // [32,64]
                  const float* __restrict__ b,    // [64]
                  float*       __restrict__ h,    // [N,64]
                  int numNodes)
{
    const int wave   = threadIdx.x >> 5;
    const int lane   = threadIdx.x & 31;
    const int tile   = blockIdx.x * 8 + wave;            // wave-uniform
    const int nTiles = (numNodes + 15) >> 4;
    if (tile >= nTiles) return;                          // uniform exit

    const int lane15 = lane & 15;
    const int kh     = lane >> 4;                        // K half-select for A/B fragments
    const int row    = tile * 16 + lane15;
    const int rowL   = min(row, numNodes - 1);           // clamp loads for ragged tail
    const bool full  = (tile * 16 + 16 <= numNodes);     // wave-uniform fast path
    const float invd = 1.0f / fmaxf(deg[rowL], 1.0f);
    const float* xr  = x   + (size_t)rowL * 32;
    const float* ar  = agg + (size_t)rowL * 32;

    #pragma unroll
    for (int ct = 0; ct < 4; ++ct) {
        const int n0 = ct * 16;
        const float bias = b[n0 + lane15];               // same col for all 8 C rows
        v8f acc = {bias, bias, bias, bias, bias, bias, bias, bias};
        #pragma unroll
        for (int k0 = 0; k0 < 32; k0 += 4) {
            const int ka = k0 + 2 * kh;
            v2f aX = *(const v2f*)(xr + ka);
            v2f aG; aG.x = ar[ka] * invd;                 aG.y = ar[ka + 1] * invd;
            v2f bL; bL.x = Wl[(size_t)ka * 64 + n0 + lane15];
                    bL.y = Wl[(size_t)(ka + 1) * 64 + n0 + lane15];
            v2f bR; bR.x = Wr[(size_t)ka * 64 + n0 + lane15];
                    bR.y = Wr[(size_t)(ka + 1) * 64 + n0 + lane15];
            acc = __builtin_amdgcn_wmma_f32_16x16x4_f32(false, aG, false, bL,
                                                        (short)0, acc, false, false);
            acc = __builtin_amdgcn_wmma_f32_16x16x4_f32(false, aX, false, bR,
                                                        (short)0, acc, false, false);
        }
        float* hp = h + (size_t)(tile * 16 + 8 * kh) * 64 + n0 + lane15;
        if (full) {                                      // unpredicated epilogue
            #pragma unroll
            for (int j = 0; j < 8; ++j)
                hp[(size_t)j * 64] = fmaxf(acc[j], 0.0f);
        } else {                                         // ragged tile (at most one)
            #pragma unroll
            for (int j = 0; j < 8; ++j)
                if (tile * 16 + j + 8 * kh < numNodes)
                    hp[(size_t)j * 64] = fmaxf(acc[j], 0.0f);
        }
    }
}

// ---------------------------------------------------------------------------
// Layer 2 node update: out = (agg2/deg) @ W_l + b + h @ W_r, K=64 -> 32
// ---------------------------------------------------------------------------
__global__ __launch_bounds__(256)
void sage_node_l2(const float* __restrict__ hs,   // [N,64]
                  const float* __restrict__ agg,  // [N,64] (sums)
                  const float* __restrict__ deg,  // [N]
                  const float* __restrict__ Wl,   // [64,32]
                  const float* __restrict__ Wr,   // [64,32]
                  const float* __restrict__ b,    // [32]
                  float*       __restrict__ out,  // [N,32]
                  int numNodes)
{
    const int wave   = threadIdx.x >> 5;
    const int lane   = threadIdx.x & 31;
    const int tile   = blockIdx.x * 8 + wave;
    const int nTiles = (numNodes + 15) >> 4;
    if (tile >= nTiles) return;

    const int lane15 = lane & 15;
    const int kh     = lane >> 4;
    const int row    = tile * 16 + lane15;
    const int rowL   = min(row, numNodes - 1);
    const bool full  = (tile * 16 + 16 <= numNodes);
    const float invd = 1.0f / fmaxf(deg[rowL], 1.0f);
    const float* hr  = hs  + (size_t)rowL * 64;
    const float* ar  = agg + (size_t)rowL * 64;

    #pragma unroll
    for (int ct = 0; ct < 2; ++ct) {
        const int n0 = ct * 16;
        const float bias = b[n0 + lane15];
        v8f acc = {bias, bias, bias, bias, bias, bias, bias, bias};
        #pragma unroll
        for (int k0 = 0; k0 < 64; k0 += 4) {
            const int ka = k0 + 2 * kh;
            v2f aH = *(const v2f*)(hr + ka);
            v2f aG; aG.x = ar[ka] * invd;                 aG.y = ar[ka + 1] * invd;
            v2f bL; bL.x = Wl[(size_t)ka * 32 + n0 + lane15];
                    bL.y = Wl[(size_t)(ka + 1) * 32 + n0 + lane15];
            v2f bR; bR.x = Wr[(size_t)ka * 32 + n0 + lane15];
                    bR.y = Wr[(size_t)(ka + 1) * 32 + n0 + lane15];
            acc = __builtin_amdgcn_wmma_f32_16x16x4_f32(false, aG, false, bL,
                                                        (short)0, acc, false, false);
            acc = __builtin_amdgcn_wmma_f32_16x16x4_f32(false, aH, false, bR,
                                                        (short)0, acc, false, false);
        }
        float* op = out + (size_t)(tile * 16 + 8 * kh) * 32 + n0 + lane15;
        if (full) {
            #pragma unroll
            for (int j = 0; j < 8; ++j)
                op[(size_t)j * 32] = acc[j];
        } else {
            #pragma unroll
            for (int j = 0; j < 8; ++j)
                if (tile * 16 + j + 8 * kh < numNodes)
                    op[(size_t)j * 32] = acc[j];
        }
    }
}

// ---------------------------------------------------------------------------
// Host-side launch. Workspace layout (floats): deg[Na] | agg1[N*32] | agg2[N*64] | h[N*64]
// Total ~64.5 MB for N=100k; first three regions zeroed each call (h is fully
// overwritten by sage_node_l1). All work issued on `stream` (graph-capturable).
// ---------------------------------------------------------------------------
extern "C" void kernel_launch(void* const* d_in, const int* in_sizes, int n_in,
                              void* d_out, int out_size, void* d_ws, size_t ws_size,
                              hipStream_t stream)
{
    const float*     x   = (const float*)d_in[0];          // [N,32]
    const long long* ei  = (const long long*)d_in[1];      // [2,E] int64 per reference
    const float*     W1l = (const float*)d_in[2];          // [32,64]
    const float*     b1  = (const float*)d_in[3];          // [64]
    const float*     W1r = (const float*)d_in[4];          // [32,64]
    const float*     W2l = (const float*)d_in[5];          // [64,32]
    const float*     b2  = (const float*)d_in[6];          // [32]
    const float*     W2r = (const float*)d_in[7];          // [64,32]

    const int       Nn = in_sizes[0] / 32;
    const long long E  = in_sizes[1] / 2;

    float* ws   = (float*)d_ws;
    size_t Na   = ((size_t)Nn + 63) & ~(size_t)63;
    float* deg  = ws;                                      // [Na]
    float* agg1 = deg  + Na;                               // [N*32]
    float* agg2 = agg1 + (size_t)Nn * 32;                  // [N*64]
    float* h    = agg2 + (size_t)Nn * 64;                  // [N*64]

    const size_t zeroBytes = (Na + (size_t)Nn * 32 + (size_t)Nn * 64) * sizeof(float);
    hipMemsetAsync(d_ws, 0, zeroBytes, stream);

    {   // layer-1 scatter (+degree)
        long long total = E << 5;
        int blocks = (int)((total + 255) / 256);
        sage_scatter<<<blocks, 256, 0, stream>>>(ei, x, agg1, deg, E, 5);
    }
    {   // layer-1 node update (fp32 WMMA) -> h
        int nTiles = (Nn + 15) / 16;
        int blocks = (nTiles + 7) / 8;
        sage_node_l1<<<blocks, 256, 0, stream>>>(x, agg1, deg, W1l, W1r, b1, h, Nn);
    }
    {   // layer-2 scatter
        long long total = E << 6;
        int blocks = (int)((total + 255) / 256);
        sage_scatter<<<blocks, 256, 0, stream>>>(ei, h, agg2, nullptr, E, 6);
    }
    {   // layer-2 node update (fp32 WMMA) -> out
        int nTiles = (Nn + 15) / 16;
        int blocks = (nTiles + 7) / 8;
        sage_node_l2<<<blocks, 256, 0, stream>>>(h, agg2, deg, W2l, W2r, b2,
                                                 (float*)d_out, Nn);
    }
}